// SpectralConv3d_82617990905915
// MI455X (gfx1250) — compile-verified
//
#include <hip/hip_runtime.h>

typedef __attribute__((ext_vector_type(2))) float v2f;
typedef __attribute__((ext_vector_type(8))) float v8f;

#define TWOPI_F 6.28318530717958647692f
#define HALFPI_F 1.57079632679489661923f
#define QPI_F 0.78539816339744830962f
#define SQRT2_F 1.41421356237309504880f

__device__ __forceinline__ int kmap16(int c) { return (c < 8) ? c : (c + 48); }

// -------- K1: partial DFT along n3 (64 -> 8 complex) via WMMA f32 16x16x4 ----
// Rows r = (b,i,n1,n2) = 524288. Output Fz[r][j]: j<8 Re(c3=j), j>=8 Im(c3=j-8).
// Fz[c3] = sum_n3 x[n3] e^{-i 2pi c3 n3/64}; Re col: cos(th), Im col: -sin(th)=cos(th+pi/2).
__global__ __launch_bounds__(256) void k1_partial_fft_z(const float* __restrict__ x,
                                                        float* __restrict__ fz) {
  const int lane = threadIdx.x & 31;
  const int wid  = threadIdx.x >> 5;
  const long rt  = (long)blockIdx.x * 8 + wid;   // row tile (32768)
  const long r0  = rt * 16;
  const int col  = lane & 15;
  const int hi   = lane >> 4;                    // 0: lanes 0-15, 1: lanes 16-31
  const int kf   = col & 7;                      // c3 for this output column
  const float phoff = (col & 8) ? HALFPI_F : 0.0f;  // Im cols: cos(th+pi/2) = -sin(th)
  const float* arow = x + (r0 + col) * 64;       // A-matrix row M = col

  v8f acc = {};
#pragma unroll
  for (int kk = 0; kk < 16; ++kk) {
    const int n0 = kk * 4 + hi * 2;
    v2f a;
    a.x = arow[n0];
    a.y = arow[n0 + 1];
    const int kn0 = (kf * n0) & 63;
    const int kn1 = (kf * (n0 + 1)) & 63;
    v2f bv;
    bv.x = __cosf(TWOPI_F * (float)kn0 * (1.0f / 64.0f) + phoff);
    bv.y = __cosf(TWOPI_F * (float)kn1 * (1.0f / 64.0f) + phoff);
    acc = __builtin_amdgcn_wmma_f32_16x16x4_f32(false, a, false, bv, (short)0, acc,
                                                false, false);
  }
#pragma unroll
  for (int jj = 0; jj < 8; ++jj) {
    const long row = r0 + jj + hi * 8;
    fz[row * 16 + col] = acc[jj];
  }
}

// -------- K2: partial DFT along n2 (64 -> 16 complex, k2 in {0..7,56..63}) ---
// Block per (b,i,n1) = 8192; 128 threads = (c2=16, c3=8).
__global__ __launch_bounds__(128) void k2_partial_fft_y(const float* __restrict__ fz,
                                                        float* __restrict__ fyz) {
  __shared__ float sFz[1024];  // [n2][16]
  __shared__ float sC[1024];   // [c2][n2]
  __shared__ float sS[1024];
  const int t = threadIdx.x;
  const long blk = blockIdx.x;
  const long base = blk * 1024;  // 64 rows * 16 floats
  for (int u = t; u < 1024; u += 128) sFz[u] = fz[base + u];
  for (int e = t; e < 1024; e += 128) {
    const int c2 = e >> 6, n2 = e & 63;
    const int kn = (kmap16(c2) * n2) & 63;
    const float th = TWOPI_F * (float)kn * (1.0f / 64.0f);
    sC[e] = __cosf(th);
    sS[e] = __sinf(th);
  }
  __syncthreads();
  const int c2 = t >> 3;
  const int c3 = t & 7;
  float re = 0.f, im = 0.f;
#pragma unroll 8
  for (int n2 = 0; n2 < 64; ++n2) {
    const float ar = sFz[n2 * 16 + c3];
    const float ai = sFz[n2 * 16 + 8 + c3];
    const float c = sC[c2 * 64 + n2];
    const float s = sS[c2 * 64 + n2];
    re += ar * c + ai * s;   // (ar+i ai)*(c - i s)
    im += ai * c - ar * s;
  }
  const long o = (blk * 16 + c2) * 8 + c3;
  fyz[o * 2] = re;
  fyz[o * 2 + 1] = im;
}

// -------- K3: partial DFT along n1 (64 -> 16 complex) ------------------------
__global__ __launch_bounds__(256) void k3_partial_fft_x(const float* __restrict__ fyz,
                                                        float* __restrict__ fx) {
  const long idx = (long)blockIdx.x * 256 + threadIdx.x;  // 262144 = (b,i,c1,c2,c3)
  const int c3 = idx & 7;
  const int c2 = (idx >> 3) & 15;
  const int c1 = (idx >> 7) & 15;
  const long bi = idx >> 11;
  const int k1 = kmap16(c1);
  float re = 0.f, im = 0.f;
  const float* src = fyz + (((bi * 64) * 16 + c2) * 8 + c3) * 2;
  for (int n1 = 0; n1 < 64; ++n1) {
    const float ar = src[0], ai = src[1];
    src += 16 * 8 * 2;
    const int kn = (k1 * n1) & 63;
    const float th = TWOPI_F * (float)kn * (1.0f / 64.0f);
    const float c = __cosf(th), s = __sinf(th);
    re += ar * c + ai * s;
    im += ai * c - ar * s;
  }
  fx[idx * 2] = re;
  fx[idx * 2 + 1] = im;
}

// -------- K4a: xh = Re - Im;  cph = cos(atan2(xhf, xh)) on the 8^3 corner ----
// xhf from conj-symmetry: F(flip x)[k] = e^{i phi} conj(F[k]), phi = 2pi(k1+k2+k3)/64
__global__ __launch_bounds__(256) void k4a_corners(const float* __restrict__ fx,
                                                   float* __restrict__ xh,
                                                   float* __restrict__ cph) {
  const long idx = (long)blockIdx.x * 256 + threadIdx.x;  // 262144
  const int c3 = idx & 7;
  const int c2 = (idx >> 3) & 15;
  const int c1 = (idx >> 7) & 15;
  const long bi = idx >> 11;
  const float a = fx[idx * 2];
  const float bI = fx[idx * 2 + 1];
  const float h = a - bI;
  xh[idx] = h;
  if (c1 < 8 && c2 < 8) {
    const int ks = (c1 + c2 + c3) & 63;
    const float ph = TWOPI_F * (float)ks * (1.0f / 64.0f);
    // cp - sp = sqrt2*cos(ph + pi/4); cp + sp = sqrt2*cos(ph - pi/4)
    const float wA = SQRT2_F * __cosf(ph + QPI_F);
    const float wB = SQRT2_F * __cosf(ph - QPI_F);
    const float hf = a * wA + bI * wB;  // Re - Im of e^{i phi} conj(F)
    const float d2 = h * h + hf * hf;
    const float cv = (d2 > 0.f) ? h * rsqrtf(d2) : 1.0f;
    cph[bi * 512 + c1 * 64 + c2 * 8 + c3] = cv;
  }
}

// -------- K4b: channel mixing OH[b,o,k] = sum_i xh[b,i,k] * w_sel[i,o,km] ----
__global__ __launch_bounds__(512) void k4b_mix(const float* __restrict__ xh,
                                               const float* __restrict__ w1,
                                               const float* __restrict__ w2,
                                               const float* __restrict__ w3,
                                               const float* __restrict__ w4,
                                               float* __restrict__ oh) {
  __shared__ float sX[512];  // [i][c3]
  const int blk = blockIdx.x;  // b*256 + c1*16 + c2  (512)
  const int c2b = blk & 15;
  const int c1b = (blk >> 4) & 15;
  const int b = blk >> 8;
  const int t = threadIdx.x;
  {
    const int i = t >> 3, c3 = t & 7;
    sX[t] = xh[((((long)b * 64 + i) * 16 + c1b) * 16 + c2b) * 8 + c3];
  }
  __syncthreads();
  const int o = t >> 3, c3 = t & 7;
  const float* w = (c1b < 8) ? ((c2b < 8) ? w1 : w3) : ((c2b < 8) ? w2 : w4);
  const int wm = ((c1b & 7) * 8 + (c2b & 7)) * 8 + c3;
  float acc = 0.f;
#pragma unroll 4
  for (int i = 0; i < 64; ++i)
    acc += sX[i * 8 + c3] * w[((long)i * 64 + o) * 512 + wm];
  oh[((((long)b * 64 + o) * 16 + c1b) * 16 + c2b) * 8 + c3] = acc;
}

// -------- K5: synthesis over k1 (16 modes -> 64 p1), complex out -------------
__global__ __launch_bounds__(256) void k5_syn_x(const float* __restrict__ oh,
                                                float* __restrict__ t1) {
  const long idx = (long)blockIdx.x * 256 + threadIdx.x;  // 1048576 = (b,o,p1,c2,c3)
  const int c3 = idx & 7;
  const int c2 = (idx >> 3) & 15;
  const int p1 = (idx >> 7) & 63;
  const long bo = idx >> 13;
  float re = 0.f, im = 0.f;
  const float* src = oh + ((bo * 16) * 16 + c2) * 8 + c3;
#pragma unroll
  for (int c1 = 0; c1 < 16; ++c1) {
    const float v = src[(long)c1 * 128];
    const int kn = (kmap16(c1) * p1) & 63;
    const float th = TWOPI_F * (float)kn * (1.0f / 64.0f);
    re += v * __cosf(th);
    im -= v * __sinf(th);
  }
  t1[idx * 2] = re;
  t1[idx * 2 + 1] = im;
}

// -------- K6: synthesis over k2 (16 modes -> 64 p2), complex -----------------
__global__ __launch_bounds__(512) void k6_syn_y(const float* __restrict__ t1,
                                                float* __restrict__ t2) {
  __shared__ float sT[256];   // [c2][c3] interleaved complex
  __shared__ float sC[1024];  // [c2][p2]
  __shared__ float sS[1024];
  const long blk = blockIdx.x;  // (b,o,p1) = 8192
  const int t = threadIdx.x;
  if (t < 256) sT[t] = t1[blk * 256 + t];
  for (int e = t; e < 1024; e += 512) {
    const int c2 = e >> 6, p2 = e & 63;
    const int kn = (kmap16(c2) * p2) & 63;
    const float th = TWOPI_F * (float)kn * (1.0f / 64.0f);
    sC[e] = __cosf(th);
    sS[e] = __sinf(th);
  }
  __syncthreads();
  const int p2 = t >> 3, c3 = t & 7;
  float re = 0.f, im = 0.f;
#pragma unroll
  for (int c2 = 0; c2 < 16; ++c2) {
    const float ar = sT[(c2 * 8 + c3) * 2];
    const float ai = sT[(c2 * 8 + c3) * 2 + 1];
    const float c = sC[c2 * 64 + p2];
    const float s = sS[c2 * 64 + p2];
    re += ar * c + ai * s;
    im += ai * c - ar * s;
  }
  const long row = blk * 64 + p2;  // (b,o,p1,p2)
  t2[row * 16 + c3] = re;
  t2[row * 16 + 8 + c3] = im;
}

// -------- K7: synthesis over k3 via WMMA + scale + cos_phase -----------------
// x_rec[p3] = sum_{k<8} Tr[k]*(cos+sin)(th) + Ti[k]*(sin-cos)(th), th=2pi k p3/64
// cos+sin = sqrt2*cos(th - pi/4); sin-cos = sqrt2*cos(th - 3pi/4).
// Selector t<8 depends only on unrolled kk (t = kk*4 + hi*2, both sub-values same
// half), so the phase offset is a compile-time constant per chunk: branch-free.
__global__ __launch_bounds__(256) void k7_syn_z_wmma(const float* __restrict__ t2,
                                                     const float* __restrict__ cph,
                                                     float* __restrict__ out) {
  const int lane = threadIdx.x & 31;
  const int wid = threadIdx.x >> 5;
  const long rt = (long)blockIdx.x * 8 + wid;  // 32768 tiles
  const long r0 = rt * 16;
  const int col = lane & 15;
  const int hi = lane >> 4;
  const int p2base = ((int)(rt & 3)) * 16;
  const int p1 = (int)((rt >> 2) & 63);
  const long bo = rt >> 8;

  const float* arow = t2 + (r0 + col) * 16;
  v2f a[4];
#pragma unroll
  for (int kk = 0; kk < 4; ++kk) {
    a[kk].x = arow[kk * 4 + hi * 2];
    a[kk].y = arow[kk * 4 + hi * 2 + 1];
  }

  v8f accs[4];
#pragma unroll
  for (int nt = 0; nt < 4; ++nt) {
    const int p3 = nt * 16 + col;
    v8f acc = {};
#pragma unroll
    for (int kk = 0; kk < 4; ++kk) {
      const int t0 = kk * 4 + hi * 2;
      const float off = (kk < 2) ? -QPI_F : -3.0f * QPI_F;  // Tr cols / Ti cols
      const int kn0 = ((t0 & 7) * p3) & 63;
      const int kn1 = (((t0 + 1) & 7) * p3) & 63;
      v2f bv;
      bv.x = SQRT2_F * __cosf(TWOPI_F * (float)kn0 * (1.0f / 64.0f) + off);
      bv.y = SQRT2_F * __cosf(TWOPI_F * (float)kn1 * (1.0f / 64.0f) + off);
      acc = __builtin_amdgcn_wmma_f32_16x16x4_f32(false, a[kk], false, bv, (short)0,
                                                  acc, false, false);
    }
    accs[nt] = acc;
  }

  const float scale = 1.0f / 262144.0f;  // 1/(64^3)
  const long cbase = bo * 512;
#pragma unroll
  for (int nt = 0; nt < 4; ++nt) {
#pragma unroll
    for (int jj = 0; jj < 8; ++jj) {
      const int m = jj + hi * 8;
      const int p2 = p2base + m;
      const int p3 = nt * 16 + col;
      float v = accs[nt][jj] * scale;
      if (p1 < 8 && p2 < 8 && p3 < 8)
        v *= cph[cbase + p1 * 64 + p2 * 8 + p3];
      out[(r0 + m) * 64 + p3] = v;
    }
  }
}

extern "C" void kernel_launch(void* const* d_in, const int* in_sizes, int n_in,
                              void* d_out, int out_size, void* d_ws, size_t ws_size,
                              hipStream_t stream) {
  (void)in_sizes; (void)n_in; (void)out_size; (void)ws_size;
  const float* x = (const float*)d_in[0];
  const float* w1 = (const float*)d_in[1];
  const float* w2 = (const float*)d_in[2];
  const float* w3 = (const float*)d_in[3];
  const float* w4 = (const float*)d_in[4];
  float* out = (float*)d_out;
  float* ws = (float*)d_ws;

  float* fz = ws;                  // 8,388,608 floats (reused as t2)
  float* fyz = fz + 8388608;       // 2,097,152 floats (reused as t1)
  float* fx = fyz + 2097152;       //   524,288 floats
  float* xh = fx + 524288;         //   262,144 floats
  float* cph = xh + 262144;        //    65,536 floats
  float* oh = cph + 65536;         //   262,144 floats  (total ~46.4 MB)
  float* t1 = fyz;                 // alias: fyz dead after K3
  float* t2 = fz;                  // alias: fz dead after K2

  k1_partial_fft_z<<<4096, 256, 0, stream>>>(x, fz);
  k2_partial_fft_y<<<8192, 128, 0, stream>>>(fz, fyz);
  k3_partial_fft_x<<<1024, 256, 0, stream>>>(fyz, fx);
  k4a_corners<<<1024, 256, 0, stream>>>(fx, xh, cph);
  k4b_mix<<<512, 512, 0, stream>>>(xh, w1, w2, w3, w4, oh);
  k5_syn_x<<<4096, 256, 0, stream>>>(oh, t1);
  k6_syn_y<<<8192, 512, 0, stream>>>(t1, t2);
  k7_syn_z_wmma<<<4096, 256, 0, stream>>>(t2, cph, out);
}